// NSVQ_50199577756057
// MI455X (gfx1250) — compile-verified
//
#include <hip/hip_runtime.h>
#include <math.h>

typedef float v2f __attribute__((ext_vector_type(2)));
typedef float v8f __attribute__((ext_vector_type(8)));

#define N_ROWS 131072
#define DIM 64
#define KCB 1024
#define ROWS_PER_BLOCK 128
#define THREADS 256
#define EPSF 1e-12f

// Dynamic LDS float layout:
// [0, 65536)        codebooks, swizzled into WMMA B-fragment order
// [65536, 66560)    ||c_k||^2
// [66560, 66688)    per-wave row min-score  (8 waves * 16 rows)
// [66688, 66816)    per-wave row argmin idx (int bits)
#define LDS_FLOATS 66816

__global__ __launch_bounds__(THREADS) void nsvq_main(
    const float* __restrict__ X, const float* __restrict__ CB,
    const float* __restrict__ RV, float* __restrict__ OUT,
    float* __restrict__ counts)
{
    extern __shared__ float lds[];
    float* cbS      = lds;            // 65536 floats
    float* cnorm    = lds + 65536;    // 1024 floats
    float* rowscore = lds + 66560;    // 128 floats
    int*   rowidx   = (int*)(lds + 66688); // 128 ints

    const int tid = threadIdx.x;

    // ---- Stage codebooks into LDS in WMMA B-fragment order ----
    // B-frag for (kt, g): lane = h*16 + kl holds CB[kt*16+kl, 4g+2h + {0,1}]
    // stored contiguously: dst = ((kt*16+g)*32 + lane)*2
    for (int idx = tid; idx < KCB * 32; idx += THREADS) {
        const int k    = idx >> 5;          // codebook row
        const int pair = idx & 31;          // which float2 of the row
        const int d    = pair * 2;
        const int g    = d >> 2;            // K-slice group (d0 = 4g)
        const int h    = (d >> 1) & 1;      // lane-half
        const int kt   = k >> 4;
        const int kl   = k & 15;
        const int lane = h * 16 + kl;
        const float2 v = *(const float2*)(CB + k * DIM + d);
        const int dst  = ((kt * 16 + g) * 32 + lane) * 2;
        cbS[dst]     = v.x;
        cbS[dst + 1] = v.y;
    }
    // ---- Codebook squared norms ----
    for (int k = tid; k < KCB; k += THREADS) {
        const float4* p = (const float4*)(CB + k * DIM);
        float s = 0.f;
        #pragma unroll
        for (int i = 0; i < 16; ++i) {
            const float4 q = p[i];
            s += q.x * q.x + q.y * q.y + q.z * q.z + q.w * q.w;
        }
        cnorm[k] = s;
    }
    __syncthreads();

    const int wave = tid >> 5;
    const int lane = tid & 31;
    const int half = lane >> 4;
    const int lrow = lane & 15;
    const int M0   = blockIdx.x * ROWS_PER_BLOCK + wave * 16;
    const int row  = M0 + lrow;

    // ---- Load A fragments (entire D=64 of this wave's 16 rows) ----
    // A-frag (16x4 f32): lane = half*16 + lrow holds X[row, 4g+2*half + {0,1}]
    v2f A[16];
    float xnp = 0.f;
    #pragma unroll
    for (int g = 0; g < 16; ++g) {
        const float2 a = *(const float2*)(X + row * DIM + g * 4 + half * 2);
        A[g][0] = a.x;
        A[g][1] = a.y;
        xnp += a.x * a.x + a.y * a.y;
    }
    const float xnorm2 = xnp + __shfl_xor(xnp, 16, 32);  // lane holds ||x_row||^2, row = M0+lrow

    float mins[8];
    int   mini[8];
    #pragma unroll
    for (int r = 0; r < 8; ++r) { mins[r] = INFINITY; mini[r] = 0; }

    // ---- Sweep all 64 codebook column tiles ----
    for (int kt = 0; kt < KCB / 16; ++kt) {
        v8f acc = {0.f, 0.f, 0.f, 0.f, 0.f, 0.f, 0.f, 0.f};
        const float* bp = cbS + kt * 1024 + lane * 2;
        #pragma unroll
        for (int g = 0; g < 16; ++g) {
            const float2 bv = *(const float2*)(bp + g * 64);  // conflict-free ds_load_b64
            v2f b;
            b[0] = bv.x;
            b[1] = bv.y;
            acc = __builtin_amdgcn_wmma_f32_16x16x4_f32(
                false, A[g], false, b, (short)0, acc, false, false);
        }
        // acc[r] = dot(x_{M0 + r + 8*half}, c_{kt*16 + lrow})
        const int   kcol = kt * 16 + lrow;
        const float cn   = cnorm[kcol];
        #pragma unroll
        for (int r = 0; r < 8; ++r) {
            const float score = cn - 2.f * acc[r];   // dist - ||x||^2
            if (score < mins[r]) { mins[r] = score; mini[r] = kcol; }
        }
    }

    // ---- Cross-lane argmin over the 16 lanes of each half ----
    #pragma unroll
    for (int r = 0; r < 8; ++r) {
        float v  = mins[r];
        int   id = mini[r];
        #pragma unroll
        for (int m = 1; m < 16; m <<= 1) {
            const float ov  = __shfl_xor(v, m, 32);
            const int   oid = __shfl_xor(id, m, 32);
            if (ov < v || (ov == v && oid < id)) { v = ov; id = oid; }
        }
        if (lrow == r) {  // one writer per row (rows r+8*half)
            rowscore[wave * 16 + r + 8 * half] = v;
            rowidx[wave * 16 + r + 8 * half]   = id;
        }
    }
    // Per-wave LDS scratch: same-wave in-order DS ops; compiler inserts s_wait_dscnt.

    // ---- Epilogue: lane handles row M0+lrow, d-range [half*32, half*32+32) ----
    const float score = rowscore[wave * 16 + lrow];
    const int   bestk = rowidx[wave * 16 + lrow];
    const float nres  = sqrtf(fmaxf(xnorm2 + score, 0.f));  // ||x - c_k*||

    const float4* rvp = (const float4*)(RV + row * DIM + half * 32);
    float4 rv[8];
    float  sr = 0.f;
    #pragma unroll
    for (int i = 0; i < 8; ++i) {
        rv[i] = rvp[i];
        sr += rv[i].x * rv[i].x + rv[i].y * rv[i].y + rv[i].z * rv[i].z + rv[i].w * rv[i].w;
    }
    const float nrand2 = sr + __shfl_xor(sr, 16, 32);
    const float scale  = nres / (sqrtf(nrand2) + EPSF);

    const float4* xp = (const float4*)(X + row * DIM + half * 32);
    float4*       op = (float4*)(OUT + row * DIM + half * 32);
    #pragma unroll
    for (int i = 0; i < 8; ++i) {
        const float4 x = xp[i];
        float4 o;
        o.x = x.x + scale * rv[i].x;
        o.y = x.y + scale * rv[i].y;
        o.z = x.z + scale * rv[i].z;
        o.w = x.w + scale * rv[i].w;
        op[i] = o;
    }
    if (half == 0) atomicAdd(&counts[bestk], 1.0f);  // exact: counts < 2^24
}

__global__ void nsvq_zero(float* __restrict__ p, int n)
{
    const int i = blockIdx.x * blockDim.x + threadIdx.x;
    if (i < n) p[i] = 0.f;
}

__global__ void nsvq_perplexity(const float* __restrict__ counts,
                                float* __restrict__ perp_out)
{
    __shared__ float red[256];
    const int t = threadIdx.x;
    float s = 0.f;
    for (int k = t; k < KCB; k += 256) {
        const float p = counts[k] * (1.0f / (float)N_ROWS);
        s += p * logf(p + EPSF);
    }
    red[t] = s;
    __syncthreads();
    for (int off = 128; off > 0; off >>= 1) {
        if (t < off) red[t] += red[t + off];
        __syncthreads();
    }
    if (t == 0) *perp_out = expf(-red[0]);
}

extern "C" void kernel_launch(void* const* d_in, const int* in_sizes, int n_in,
                              void* d_out, int out_size, void* d_ws, size_t ws_size,
                              hipStream_t stream)
{
    (void)in_sizes; (void)n_in; (void)out_size; (void)d_ws; (void)ws_size;
    const float* X  = (const float*)d_in[0];   // input_data   (N, D)
    const float* CB = (const float*)d_in[1];   // codebooks    (K, D)
    const float* RV = (const float*)d_in[2];   // random_vector(N, D)

    float* OUT    = (float*)d_out;                       // quantized (N*D)
    float* perp   = OUT + (size_t)N_ROWS * DIM;          // 1 float
    float* counts = perp + 1;                            // K floats

    nsvq_zero<<<(KCB + 1 + 255) / 256, 256, 0, stream>>>(perp, KCB + 1);
    nsvq_main<<<N_ROWS / ROWS_PER_BLOCK, THREADS, LDS_FLOATS * sizeof(float), stream>>>(
        X, CB, RV, OUT, counts);
    nsvq_perplexity<<<1, 256, 0, stream>>>(counts, perp);
}